// BiLstmCellLayer_9792525435316
// MI455X (gfx1250) — compile-verified
//
#include <hip/hip_runtime.h>
#include <math.h>

typedef __attribute__((ext_vector_type(2))) float v2f;
typedef __attribute__((ext_vector_type(8))) float v8f;

constexpr int B = 64, S = 512, I = 1024, H = 1024;
constexpr int NC  = 16;        // h-columns per workgroup
constexpr int WGX = H / NC;    // 64 workgroups per direction
constexpr int KC  = 32;        // K staging chunk
constexpr int NCHUNK = I / KC; // 32 chunks per matrix
constexpr int LDA = 36;        // padded LDS stride for A/B stages (16B-aligned rows)
constexpr int LDG = 68;        // padded LDS stride for gate tile

// Async global->LDS DMA (cache->LDS, no VGPR round trip, tracked by ASYNCcnt).
// LDS operand: low 32 bits of the generic pointer == wave-relative LDS byte addr.
__device__ __forceinline__ void async_ld16(const float* g, float* l) {
  unsigned la = (unsigned)(size_t)l;
  asm volatile("global_load_async_to_lds_b128 %0, %1, off"
               :: "v"(la), "v"(g) : "memory");
}
__device__ __forceinline__ void wait_async_le4() {
  asm volatile("s_wait_asynccnt 0x4" ::: "memory");
}
__device__ __forceinline__ void wait_async_0() {
  asm volatile("s_wait_asynccnt 0x0" ::: "memory");
}

__device__ __forceinline__ void gridBarrier(unsigned* cnt, unsigned* gen, unsigned n) {
  __syncthreads();
  if (threadIdx.x == 0) {
    __threadfence();  // release h writes to device scope
    unsigned g = __hip_atomic_load(gen, __ATOMIC_ACQUIRE, __HIP_MEMORY_SCOPE_AGENT);
    unsigned v = __hip_atomic_fetch_add(cnt, 1u, __ATOMIC_ACQ_REL, __HIP_MEMORY_SCOPE_AGENT);
    if (v == n - 1u) {
      __hip_atomic_store(cnt, 0u, __ATOMIC_RELAXED, __HIP_MEMORY_SCOPE_AGENT);
      __hip_atomic_store(gen, g + 1u, __ATOMIC_RELEASE, __HIP_MEMORY_SCOPE_AGENT);
    } else {
      while (__hip_atomic_load(gen, __ATOMIC_ACQUIRE, __HIP_MEMORY_SCOPE_AGENT) == g) {
        __builtin_amdgcn_s_sleep(2);
      }
    }
  }
  __syncthreads();
}

__global__ __launch_bounds__(256, 1)
void bilstm_scan_kernel(const float* __restrict__ xf, const float* __restrict__ xb,
                        const float* __restrict__ Wih_f, const float* __restrict__ Whh_f,
                        const float* __restrict__ bih_f, const float* __restrict__ bhh_f,
                        const float* __restrict__ Wih_b, const float* __restrict__ Whh_b,
                        const float* __restrict__ bih_b, const float* __restrict__ bhh_b,
                        float* __restrict__ out, float* __restrict__ ws) {
  __shared__ alignas(16) float ldsA[2][64 * LDA];  // activation chunks (double buffer)
  __shared__ alignas(16) float ldsB[2][64 * LDA];  // weight chunks (double buffer)
  __shared__ alignas(16) float ldsG[64 * LDG];     // gate tile: 64 x (4 gates * 16)

  const int d    = blockIdx.y;            // 0 = forward, 1 = backward
  const int cb   = blockIdx.x * NC;       // h-column base
  const int tid  = threadIdx.x;
  const int w    = tid >> 5;              // wave id (wave32)
  const int lane = tid & 31;
  const int lh   = lane & 15;
  const int kh   = lane >> 4;             // 0 or 1 (K-half of fragment)
  const int rt   = w & 3;                 // row tile (16 batch rows)
  const int g0   = w >> 2;                // gates handled: {g0, g0+2}
  const int g1   = g0 + 2;

  const float* x   = d ? xb    : xf;
  const float* Wih = d ? Wih_b : Wih_f;
  const float* Whh = d ? Whh_b : Whh_f;
  const float* bih = d ? bih_b : bih_f;
  const float* bhh = d ? bhh_b : bhh_f;
  float* outd = out + (size_t)d * B * S * H;

  unsigned* cnt = (unsigned*)ws + 2 * d;
  unsigned* gen = (unsigned*)ws + 2 * d + 1;
  float* hbuf = ws + 64 + (size_t)d * 2 * B * H;   // double-buffered h, 256B offset

  // Per-lane bias for this column (C-layout: every VGPR of a lane is the same N)
  const float bs0 = bih[g0 * H + cb + lh] + bhh[g0 * H + cb + lh];
  const float bs1 = bih[g1 * H + cb + lh] + bhh[g1 * H + cb + lh];

  float cc[4] = {0.f, 0.f, 0.f, 0.f};    // cell state lives in registers

  for (int t = 0; t < S; ++t) {
    const int tx = d ? (S - 1 - t) : t;
    const float* hr = hbuf + (size_t)(t & 1) * (B * H);
    float*       hw = hbuf + (size_t)((t + 1) & 1) * (B * H);

    v8f acc0, acc1;
    #pragma unroll
    for (int v = 0; v < 8; ++v) { acc0[v] = bs0; acc1[v] = bs1; }

    const int nmat = (t == 0) ? 1 : 2;   // h == 0 at t=0 -> skip recurrent GEMM
    for (int m = 0; m < nmat; ++m) {
      const float* Wsrc = m ? Whh : Wih; // both have row stride 1024

      // Issue async DMA of one 64x32 A-chunk + 64x32 B-chunk (4 ops per thread)
      auto stage = [&](int k0, int pb) {
        #pragma unroll
        for (int i = 0; i < 2; ++i) {
          int idx4 = tid + i * 256;
          int r  = idx4 >> 3;
          int k4 = (idx4 & 7) << 2;
          const float* src = m ? (hr + (size_t)r * H + k0 + k4)
                               : (x + ((size_t)(r * S + tx)) * I + k0 + k4);
          async_ld16(src, &ldsA[pb][r * LDA + k4]);
        }
        #pragma unroll
        for (int i = 0; i < 2; ++i) {
          int idx4 = tid + i * 256;
          int c  = idx4 >> 3;
          int k4 = (idx4 & 7) << 2;
          int grow = (c >> 4) * H + cb + (c & 15);
          async_ld16(Wsrc + (size_t)grow * 1024 + k0 + k4,
                     &ldsB[pb][c * LDA + k4]);
        }
      };

      stage(0, 0);                        // prologue: chunk 0 in flight
      for (int ci = 0; ci < NCHUNK; ++ci) {
        const int pb = ci & 1;
        if (ci + 1 < NCHUNK) {
          stage((ci + 1) * KC, pb ^ 1);   // overlap next DMA with this compute
          wait_async_le4();               // oldest 4 (chunk ci) retired, in-order
        } else {
          wait_async_0();
        }
        __syncthreads();                  // chunk ci visible to all waves

        #pragma unroll
        for (int kk = 0; kk < KC; kk += 4) {
          v2f a  = *(const v2f*)&ldsA[pb][(rt * 16 + lh) * LDA + kk + kh * 2];
          v2f b0 = *(const v2f*)&ldsB[pb][(g0 * 16 + lh) * LDA + kk + kh * 2];
          v2f b1 = *(const v2f*)&ldsB[pb][(g1 * 16 + lh) * LDA + kk + kh * 2];
          acc0 = __builtin_amdgcn_wmma_f32_16x16x4_f32(
              false, a, false, b0, (short)0, acc0, false, false);
          acc1 = __builtin_amdgcn_wmma_f32_16x16x4_f32(
              false, a, false, b1, (short)0, acc1, false, false);
        }
        __syncthreads();                  // buf[pb] free before it is re-written
      }
    }

    // Dump gate tile to LDS so each thread can combine i/f/g/o for its element
    #pragma unroll
    for (int v = 0; v < 8; ++v) {
      int row = rt * 16 + v + kh * 8;     // C layout: lanes 16-31 hold M = v + 8
      ldsG[row * LDG + g0 * 16 + lh] = acc0[v];
      ldsG[row * LDG + g1 * 16 + lh] = acc1[v];
    }
    __syncthreads();

    // Pointwise LSTM update; c stays in registers, h -> double buffer + output
    #pragma unroll
    for (int i = 0; i < 4; ++i) {
      int idx = i * 256 + tid;
      int row = idx >> 4;
      int col = idx & 15;
      float ig = ldsG[row * LDG +      col];
      float fg = ldsG[row * LDG + 16 + col];
      float gg = ldsG[row * LDG + 32 + col];
      float og = ldsG[row * LDG + 48 + col];
      float si = 1.f / (1.f + __expf(-ig));
      float sf = 1.f / (1.f + __expf(-fg));
      float so = 1.f / (1.f + __expf(-og));
      float gt = tanhf(gg);
      float c  = sf * cc[i] + si * gt;
      cc[i] = c;
      float h = so * tanhf(c);
      hw[(size_t)row * H + cb + col] = h;
      outd[((size_t)(row * S + tx)) * H + cb + col] = h;
    }

    // All 64 workgroups of this direction must see the new h before step t+1
    gridBarrier(cnt, gen, (unsigned)WGX);
  }
}

extern "C" void kernel_launch(void* const* d_in, const int* in_sizes, int n_in,
                              void* d_out, int out_size, void* d_ws, size_t ws_size,
                              hipStream_t stream) {
  (void)in_sizes; (void)n_in; (void)out_size; (void)ws_size;
  // zero the 4 barrier words (graph-capture-safe async memset)
  hipMemsetAsync(d_ws, 0, 4 * sizeof(unsigned), stream);

  const float* xf    = (const float*)d_in[0];
  const float* xb    = (const float*)d_in[1];
  const float* Wih_f = (const float*)d_in[2];
  const float* Whh_f = (const float*)d_in[3];
  const float* bih_f = (const float*)d_in[4];
  const float* bhh_f = (const float*)d_in[5];
  const float* Wih_b = (const float*)d_in[6];
  const float* Whh_b = (const float*)d_in[7];
  const float* bih_b = (const float*)d_in[8];
  const float* bhh_b = (const float*)d_in[9];

  dim3 grid(WGX, 2), block(256);
  bilstm_scan_kernel<<<grid, block, 0, stream>>>(
      xf, xb, Wih_f, Whh_f, bih_f, bhh_f, Wih_b, Whh_b, bih_b, bhh_b,
      (float*)d_out, (float*)d_ws);
}